// GCN_74912819577633
// MI455X (gfx1250) — compile-verified
//
#include <hip/hip_runtime.h>
#include <hip/hip_bf16.h>
#include <math.h>

typedef __attribute__((ext_vector_type(2))) float v2f;
typedef __attribute__((ext_vector_type(8))) float v8f;

#define N_NODES 100000
#define N_EDGES 1600000
#define N_GRAPH 256
#define HDIM    64
#define CDIM    10

// ---------------------------------------------------------------------------
// init: deg = 1.0 (self loop weight), pool region (cnt + sums + maxes) = 0
// ---------------------------------------------------------------------------
__global__ void k_init(float* __restrict__ deg, float* __restrict__ pool,
                       int n, int poolCount) {
    int t = blockIdx.x * blockDim.x + threadIdx.x;
    if (t < n) deg[t] = 1.0f;
    if (t < poolCount) pool[t] = 0.0f;
}

// deg[dst] += 1 per edge
__global__ void k_degree(const int* __restrict__ dst, float* __restrict__ deg, int E) {
    int e = blockIdx.x * blockDim.x + threadIdx.x;
    if (e < E) atomicAdd(&deg[dst[e]], 1.0f);
}

// deg -> dis = rsqrt(deg) in place; also count nodes per graph
__global__ void k_finish_deg(float* __restrict__ deg, const int* __restrict__ batch,
                             float* __restrict__ cnt, int n) {
    int i = blockIdx.x * blockDim.x + threadIdx.x;
    if (i >= n) return;
    deg[i] = rsqrtf(deg[i]);
    atomicAdd(&cnt[batch[i]], 1.0f);
}

// ---------------------------------------------------------------------------
// GEMM: Y[N x 64] = X[N x 64] @ W[64 x 64] with V_WMMA_F32_16X16X4_F32.
// 256 threads = 8 waves; each wave owns one 16x16 output tile:
// 2 row-tiles x 4 col-tiles per block -> 32 rows/block. N % 32 == 0 so EXEC
// is all-ones everywhere (WMMA requirement).
// ---------------------------------------------------------------------------
__global__ __launch_bounds__(256)
void k_gemm64(const float* __restrict__ X, const float* __restrict__ W,
              float* __restrict__ Y) {
    const int wave = threadIdx.x >> 5;           // 0..7
    const int lane = threadIdx.x & 31;
    const int r0 = (blockIdx.x * 2 + (wave >> 2)) * 16;   // row tile base
    const int c0 = (wave & 3) * 16;                       // col tile base
    const int ln    = lane & 15;                 // row (A) / col (B,C)
    const int khalf = (lane >> 4) << 1;          // 0 or 2
    const float* xrow = X + (size_t)(r0 + ln) * HDIM;

    v8f acc = {};
#pragma unroll
    for (int kk = 0; kk < HDIM; kk += 4) {
        v2f a, b;
        a.x = xrow[kk + khalf];
        a.y = xrow[kk + khalf + 1];
        b.x = W[(kk + khalf)     * HDIM + c0 + ln];
        b.y = W[(kk + khalf + 1) * HDIM + c0 + ln];
        acc = __builtin_amdgcn_wmma_f32_16x16x4_f32(
            /*neg_a=*/false, a, /*neg_b=*/false, b,
            /*c_mod=*/(short)0, acc, /*reuse_a=*/false, /*reuse_b=*/false);
    }
    const int mhi = lane >> 4;                   // 0/1 -> rows 0-7 / 8-15
#pragma unroll
    for (int v = 0; v < 8; ++v) {
        int row = r0 + v + 8 * mhi;
        Y[(size_t)row * HDIM + c0 + ln] = acc[v];
    }
}

// ---------------------------------------------------------------------------
// agg init with self-loop term: agg[i,:] = hw[i,:] * dis[i]^2
// 16 threads per node, float4 each.
// ---------------------------------------------------------------------------
__global__ void k_selfloop(const float* __restrict__ hw, const float* __restrict__ dis,
                           float* __restrict__ agg, int n) {
    int t = blockIdx.x * blockDim.x + threadIdx.x;
    int i = t >> 4;
    if (i >= n) return;
    int f0 = (t & 15) * 4;
    float d = dis[i];
    float w = d * d;
    float4 v = *(const float4*)(hw + (size_t)i * HDIM + f0);
    v.x *= w; v.y *= w; v.z *= w; v.w *= w;
    *(float4*)(agg + (size_t)i * HDIM + f0) = v;
}

// ---------------------------------------------------------------------------
// edge scatter-add: agg[dst,:] += hw[src,:] * dis[src]*dis[dst]
// 16 threads/edge -> coalesced 256B gather per edge; h (25.6MB) lives in L2.
// ---------------------------------------------------------------------------
__global__ void k_edge_agg(const float* __restrict__ hw,
                           const int* __restrict__ src, const int* __restrict__ dst,
                           const float* __restrict__ dis,
                           float* __restrict__ agg, int E) {
    int t = blockIdx.x * blockDim.x + threadIdx.x;
    int e = t >> 4;
    if (e >= E) return;
    int f0 = (t & 15) * 4;
    int s = src[e];
    int d = dst[e];
    float norm = dis[s] * dis[d];
    float4 v = *(const float4*)(hw + (size_t)s * HDIM + f0);
    float* out = agg + (size_t)d * HDIM + f0;
    atomicAdd(out + 0, v.x * norm);
    atomicAdd(out + 1, v.y * norm);
    atomicAdd(out + 2, v.z * norm);
    atomicAdd(out + 3, v.w * norm);
}

// ---------------------------------------------------------------------------
// h = relu(agg + b) in place, then pool: sum + max per graph.
// post-ReLU values are >= 0, so u32 atomicMax on float bits is order-correct
// and the 0-init already realizes jnp.maximum(segment_max, 0).
// ---------------------------------------------------------------------------
__global__ void k_bias_relu_pool(float* __restrict__ h, const float* __restrict__ bias,
                                 const int* __restrict__ batch,
                                 float* __restrict__ s, unsigned* __restrict__ mx, int n) {
    int t = blockIdx.x * blockDim.x + threadIdx.x;
    int i = t >> 6;
    if (i >= n) return;
    int f = t & 63;
    float v = h[(size_t)i * HDIM + f] + bias[f];
    v = fmaxf(v, 0.0f);
    h[(size_t)i * HDIM + f] = v;
    int g = batch[i];
    atomicAdd(&s[g * HDIM + f], v);
    atomicMax(&mx[g * HDIM + f], __float_as_uint(v));
}

// ---------------------------------------------------------------------------
// z[g] = concat(mean, max, sum) summed over the 3 layers
// ---------------------------------------------------------------------------
__global__ void k_combine(const float* __restrict__ cnt,
                          const float* __restrict__ s1, const float* __restrict__ s2,
                          const float* __restrict__ s3,
                          const unsigned* __restrict__ m1, const unsigned* __restrict__ m2,
                          const unsigned* __restrict__ m3,
                          float* __restrict__ z) {
    int t = blockIdx.x * blockDim.x + threadIdx.x;   // G*64 threads
    if (t >= N_GRAPH * HDIM) return;
    int g = t >> 6;
    int f = t & 63;
    float c = fmaxf(cnt[g], 1.0f);
    float ssum = s1[t] + s2[t] + s3[t];
    float msum = __uint_as_float(m1[t]) + __uint_as_float(m2[t]) + __uint_as_float(m3[t]);
    float* zr = z + (size_t)g * (3 * HDIM);
    zr[f]            = ssum / c;
    zr[HDIM + f]     = msum;
    zr[2 * HDIM + f] = ssum;
}

// ---------------------------------------------------------------------------
// MLP head + log_softmax; one 64-thread block per graph, LDS staging.
// ---------------------------------------------------------------------------
__global__ __launch_bounds__(64)
void k_mlp(const float* __restrict__ z,
           const float* __restrict__ w1, const float* __restrict__ b1,
           const float* __restrict__ w2, const float* __restrict__ b2,
           const float* __restrict__ w3, const float* __restrict__ b3,
           float* __restrict__ out) {
    __shared__ float zr[3 * HDIM];
    __shared__ float a1[HDIM];
    __shared__ float a2[HDIM / 2];
    __shared__ float lg[CDIM];
    __shared__ float lse;
    int g = blockIdx.x, t = threadIdx.x;
    zr[t]            = z[(size_t)g * 192 + t];
    zr[t + 64]       = z[(size_t)g * 192 + 64 + t];
    zr[t + 128]      = z[(size_t)g * 192 + 128 + t];
    __syncthreads();
    float acc = b1[t];
#pragma unroll 4
    for (int k = 0; k < 192; ++k) acc += zr[k] * w1[k * 64 + t];
    a1[t] = fmaxf(acc, 0.0f);
    __syncthreads();
    if (t < 32) {
        float a = b2[t];
#pragma unroll 4
        for (int k = 0; k < 64; ++k) a += a1[k] * w2[k * 32 + t];
        a2[t] = fmaxf(a, 0.0f);
    }
    __syncthreads();
    if (t < CDIM) {
        float a = b3[t];
#pragma unroll
        for (int k = 0; k < 32; ++k) a += a2[k] * w3[k * CDIM + t];
        lg[t] = a;
    }
    __syncthreads();
    if (t == 0) {
        float m = lg[0];
        for (int c = 1; c < CDIM; ++c) m = fmaxf(m, lg[c]);
        float se = 0.0f;
        for (int c = 0; c < CDIM; ++c) se += expf(lg[c] - m);
        lse = m + logf(se);
    }
    __syncthreads();
    if (t < CDIM) out[(size_t)g * CDIM + t] = lg[t] - lse;
}

// ---------------------------------------------------------------------------
extern "C" void kernel_launch(void* const* d_in, const int* in_sizes, int n_in,
                              void* d_out, int out_size, void* d_ws, size_t ws_size,
                              hipStream_t stream) {
    const float* x     = (const float*)d_in[0];
    const int*   ei    = (const int*)d_in[1];
    const int*   batch = (const int*)d_in[2];
    const float* W1 = (const float*)d_in[3];  const float* b1 = (const float*)d_in[4];
    const float* W2 = (const float*)d_in[5];  const float* b2 = (const float*)d_in[6];
    const float* W3 = (const float*)d_in[7];  const float* b3 = (const float*)d_in[8];
    const float* fc1w = (const float*)d_in[9];  const float* fc1b = (const float*)d_in[10];
    const float* fc2w = (const float*)d_in[11]; const float* fc2b = (const float*)d_in[12];
    const float* fc3w = (const float*)d_in[13]; const float* fc3b = (const float*)d_in[14];
    float* out = (float*)d_out;

    const int* src = ei;
    const int* dst = ei + N_EDGES;

    // workspace layout (floats)
    float* ws   = (float*)d_ws;
    float* dis  = ws;                                  // N (deg -> dis in place)
    float* hw   = dis + N_NODES;                       // N*64 (GEMM output, pre-agg)
    float* bufA = hw + (size_t)N_NODES * HDIM;         // N*64
    float* bufB = bufA + (size_t)N_NODES * HDIM;       // N*64
    float* cnt  = bufB + (size_t)N_NODES * HDIM;       // G
    float* s1   = cnt + N_GRAPH;                       // G*64 each
    float* s2   = s1 + N_GRAPH * HDIM;
    float* s3   = s2 + N_GRAPH * HDIM;
    unsigned* m1 = (unsigned*)(s3 + N_GRAPH * HDIM);   // G*64 each (float bits)
    unsigned* m2 = m1 + N_GRAPH * HDIM;
    unsigned* m3 = m2 + N_GRAPH * HDIM;
    float* z    = (float*)(m3 + N_GRAPH * HDIM);       // G*192

    const int poolCount = N_GRAPH + 6 * N_GRAPH * HDIM; // cnt + 3 sums + 3 maxes
    const int B = 256;
    auto gb = [](long long n, int b) { return (unsigned)((n + b - 1) / b); };

    // degrees + per-graph counts
    k_init<<<gb(N_NODES, B), B, 0, stream>>>(dis, cnt, N_NODES, poolCount);
    k_degree<<<gb(N_EDGES, B), B, 0, stream>>>(dst, dis, N_EDGES);
    k_finish_deg<<<gb(N_NODES, B), B, 0, stream>>>(dis, batch, cnt, N_NODES);

    const float* layer_in[3] = { x, bufA, bufB };
    float*       layer_out[3] = { bufA, bufB, bufA };
    const float* Ws[3] = { W1, W2, W3 };
    const float* bs[3] = { b1, b2, b3 };
    float*       sl[3] = { s1, s2, s3 };
    unsigned*    ml[3] = { m1, m2, m3 };

    for (int l = 0; l < 3; ++l) {
        k_gemm64<<<N_NODES / 32, B, 0, stream>>>(layer_in[l], Ws[l], hw);
        k_selfloop<<<gb((long long)N_NODES * 16, B), B, 0, stream>>>(hw, dis, layer_out[l], N_NODES);
        k_edge_agg<<<gb((long long)N_EDGES * 16, B), B, 0, stream>>>(hw, src, dst, dis,
                                                                     layer_out[l], N_EDGES);
        k_bias_relu_pool<<<gb((long long)N_NODES * HDIM, B), B, 0, stream>>>(
            layer_out[l], bs[l], batch, sl[l], ml[l], N_NODES);
    }

    k_combine<<<gb(N_GRAPH * HDIM, B), B, 0, stream>>>(cnt, s1, s2, s3, m1, m2, m3, z);
    k_mlp<<<N_GRAPH, 64, 0, stream>>>(z, fc1w, fc1b, fc2w, fc2b, fc3w, fc3b, out);
}